// DifferentiableMPC_72052371358247
// MI455X (gfx1250) — compile-verified
//
#include <hip/hip_runtime.h>
#include <stdint.h>

#define TPB 128
#define HOR 50

typedef unsigned int v4u __attribute__((ext_vector_type(4)));
typedef int          v4i __attribute__((ext_vector_type(4)));
typedef int          v8i __attribute__((ext_vector_type(8)));

#if defined(__has_builtin)
#if __has_builtin(__builtin_amdgcn_tensor_load_to_lds) && __has_builtin(__builtin_amdgcn_s_wait_tensorcnt)
#define USE_TDM 1
#endif
#if __has_builtin(__builtin_amdgcn_fmed3f)
#define HAVE_MED3 1
#endif
#endif

__device__ __forceinline__ float clampf(float x, float lo, float hi) {
#if defined(HAVE_MED3)
    return __builtin_amdgcn_fmed3f(x, lo, hi);   // single v_med3_f32
#else
    return __builtin_fminf(__builtin_fmaxf(x, lo), hi);
#endif
}

#if defined(USE_TDM)
// 1-D TDM copy: nelem f32 elements from global to LDS byte-offset lds_off.
// D# group0: count=1, lds_addr, 57b global addr, type=2.
// D# group1: data_size=4B, tensor_dim0=nelem, tensor_dim1=1, tile_dim0=nelem,
//            tile_dim1=0 (1-D tile), tensor_dim0_stride=nelem.
__device__ __forceinline__ void tdm_copy_1d_f32(const void* gsrc, unsigned lds_off, unsigned nelem) {
    unsigned long long ga = (unsigned long long)(uintptr_t)gsrc;
    v4u g0;
    g0.x = 1u;                                                   // count=1 (valid)
    g0.y = lds_off;                                              // lds_addr
    g0.z = (unsigned)ga;                                         // global_addr[31:0]
    g0.w = (unsigned)((ga >> 32) & 0x01FFFFFFull) | 0x80000000u; // global_addr[56:32] | type=2
    v8i g1;
    g1.s0 = (int)(2u << 16);                     // data_size = 2 (4 bytes)
    g1.s1 = (int)(nelem << 16);                  // tensor_dim0[15:0] @ bits 63:48
    g1.s2 = (int)((nelem >> 16) | (1u << 16));   // tensor_dim0[31:16] | tensor_dim1=1
    g1.s3 = (int)(nelem << 16);                  // tile_dim0 @ bits 127:112
    g1.s4 = 0;                                   // tile_dim1=0, tile_dim2=0
    g1.s5 = (int)nelem;                          // tensor_dim0_stride[31:0]
    g1.s6 = 0;
    g1.s7 = 0;
    v4i z4 = {0, 0, 0, 0};
    v8i z8 = {0, 0, 0, 0, 0, 0, 0, 0};
    __builtin_amdgcn_tensor_load_to_lds(g0, g1, z4, z4, z8, 0);
}
#endif

__global__ __launch_bounds__(TPB) void mpc_kernel(const float* __restrict__ state,
                                                  const float* __restrict__ goal,
                                                  float* __restrict__ out,
                                                  int B) {
    constexpr float DT = 0.1f, DAMP_V = 0.95f, DAMP_W = 0.9f;
    constexpr float MAXV = 2.0f, MAXW = 2.0f, MAXAV = 1.0f, MAXAW = 2.0f;
    constexpr float LR = 0.5f;
    constexpr float QP = 10.0f, QV = 5.0f, QTH = 2.0f, RC = 0.1f;
    constexpr float QFP = 50.0f, QFV = 10.0f, QFTH = 5.0f;
    constexpr int NUM_ITERS = 15;

    __shared__ float st_stage[TPB * 5];   // staged state rows for this block
    __shared__ float gl_stage[TPB * 4];   // staged goal rows

    const int tid = threadIdx.x;
    const int b   = blockIdx.x * TPB + tid;

    // ---- Stage this block's inputs into LDS (TDM DMA if available) ----
#if defined(USE_TDM)
    if (tid == 0) {
        tdm_copy_1d_f32(state + (size_t)blockIdx.x * (TPB * 5),
                        (unsigned)(uintptr_t)st_stage, TPB * 5);
        tdm_copy_1d_f32(goal + (size_t)blockIdx.x * (TPB * 4),
                        (unsigned)(uintptr_t)gl_stage, TPB * 4);
        __builtin_amdgcn_s_wait_tensorcnt(0);
    }
    __syncthreads();
#else
    for (int i = tid; i < TPB * 5; i += TPB)
        st_stage[i] = state[(size_t)blockIdx.x * (TPB * 5) + i];
    for (int i = tid; i < TPB * 4; i += TPB)
        gl_stage[i] = goal[(size_t)blockIdx.x * (TPB * 4) + i];
    __syncthreads();
#endif

    const float s0x = st_stage[tid * 5 + 0];
    const float s0y = st_stage[tid * 5 + 1];
    const float s0t = st_stage[tid * 5 + 2];
    const float s0v = st_stage[tid * 5 + 3];
    const float s0w = st_stage[tid * 5 + 4];
    const float gx  = gl_stage[tid * 4 + 0];
    const float gy  = gl_stage[tid * 4 + 1];
    const float gv  = gl_stage[tid * 4 + 2];
    const float gth = gl_stage[tid * 4 + 3];

    // All per-step history lives in registers (loops fully unrolled -> static
    // indices; occupancy is problem-limited at ~2 waves/SIMD, so VGPRs are free).
    float av[HOR], aw[HOR];      // action sequence
    float hv[HOR], hw[HOR];      // forward v_t / w_t (inputs of step t)
    float hc[HOR], hs[HOR];      // cos(th_t) / sin(th_t) cached for backward
#pragma unroll
    for (int t = 0; t < HOR; ++t) { av[t] = 0.0f; aw[t] = 0.0f; }

#pragma unroll 1
    for (int it = 0; it < NUM_ITERS; ++it) {
        // ---------------- forward rollout, record history ----------------
        float x = s0x, y = s0y, th = s0t, v = s0v, w = s0w;
#pragma unroll
        for (int t = 0; t < HOR; ++t) {
            hv[t] = v;
            hw[t] = w;
            v = clampf(v * DAMP_V + av[t] * DT, -MAXV, MAXV);
            w = clampf(w * DAMP_W + aw[t] * DT, -MAXW, MAXW);
            float c = __cosf(th);
            float s = __sinf(th);
            hc[t] = c;
            hs[t] = s;
            x += v * c * DT;
            y += v * s * DT;
            th += w * DT;
        }

        // ---------------- adjoint (backward) sweep + fused action update ----
        // lambda = dC/ds_{t+1}; start with terminal cost at s_50
        float lx  = 2.0f * QFP * (x - gx);
        float ly  = 2.0f * QFP * (y - gy);
        float lth = 2.0f * QFTH * (th - gth);
        float lv  = 2.0f * QFV * (v - gv);
        float lw  = 0.0f;
        float v1 = v, w1 = w;  // v_{t+1}, w_{t+1}
#pragma unroll
        for (int t = HOR - 1; t >= 0; --t) {
            // cached trig of th_t; reconstruct s_t position/heading in reverse
            float c = hc[t];
            float s = hs[t];
            float th_t = th - w1 * DT;
            float x_t = x - v1 * c * DT;
            float y_t = y - v1 * s * DT;
            float v0t = hv[t];
            float w0t = hw[t];

            float gthv = lth + (ly * c - lx * s) * v1 * DT;   // d/d th_t
            float gvp  = lv + (lx * c + ly * s) * DT;         // d/d v'
            float gwp  = lw + lth * DT;                        // d/d w'

            float pv = v0t * DAMP_V + av[t] * DT;              // pre-clip values
            float pw = w0t * DAMP_W + aw[t] * DT;
            float gpv = (__builtin_fabsf(pv) < MAXV) ? gvp : 0.0f;
            float gpw = (__builtin_fabsf(pw) < MAXW) ? gwp : 0.0f;

            // total gradient wrt a_t (ctrl term + dynamics term); update in place
            float ga_v = gpv * DT + 2.0f * RC * av[t];
            float ga_w = gpw * DT + 2.0f * RC * aw[t];
            av[t] = clampf(av[t] - LR * ga_v, -MAXAV, MAXAV);
            aw[t] = clampf(aw[t] - LR * ga_w, -MAXAW, MAXAW);

            // lambda at s_t (dynamics part)
            lth = gthv;
            lv  = gpv * DAMP_V;
            lw  = gpw * DAMP_W;
            // stage cost gradient at mid states s_1..s_49
            if (t >= 1) {
                lx  += 2.0f * QP * (x_t - gx);
                ly  += 2.0f * QP * (y_t - gy);
                lth += 2.0f * QTH * (th_t - gth);
                lv  += 2.0f * QV * (v0t - gv);
            }
            x = x_t; y = y_t; th = th_t;
            v1 = v0t; w1 = w0t;
        }
    }

    // ---------------- final rollout + cost ----------------
    float x = s0x, y = s0y, th = s0t, v = s0v, w = s0w;
    float cost = 0.0f;
#pragma unroll
    for (int t = 0; t < HOR; ++t) {
        v = clampf(v * DAMP_V + av[t] * DT, -MAXV, MAXV);
        w = clampf(w * DAMP_W + aw[t] * DT, -MAXW, MAXW);
        float c = __cosf(th);
        float s = __sinf(th);
        x += v * c * DT;
        y += v * s * DT;
        th += w * DT;
        cost += RC * (av[t] * av[t] + aw[t] * aw[t]);
        if (t < HOR - 1) {  // mid states s_1..s_49
            float ex = x - gx, ey = y - gy, ev = v - gv, et = th - gth;
            cost += QP * (ex * ex + ey * ey) + QV * ev * ev + QTH * et * et;
        }
    }
    {
        float ex = x - gx, ey = y - gy, ev = v - gv, et = th - gth;
        cost += QFP * (ex * ex + ey * ey) + QFV * ev * ev + QFTH * et * et;
    }

    if (b < B) {
        reinterpret_cast<float2*>(out)[b] = make_float2(av[0], aw[0]);  // actions[:,0,:]
        out[(size_t)2 * B + b] = cost;                                   // final_cost
    }
}

extern "C" void kernel_launch(void* const* d_in, const int* in_sizes, int n_in,
                              void* d_out, int out_size, void* d_ws, size_t ws_size,
                              hipStream_t stream) {
    const float* state = (const float*)d_in[0];
    const float* goal  = (const float*)d_in[1];
    float* out = (float*)d_out;
    const int B = in_sizes[0] / 5;
    const int grid = (B + TPB - 1) / TPB;
    hipLaunchKernelGGL(mpc_kernel, dim3(grid), dim3(TPB), 0, stream, state, goal, out, B);
}